// DenoiseEncoder_36077725286625
// MI455X (gfx1250) — compile-verified
//
#include <hip/hip_runtime.h>
#include <hip/hip_bf16.h>
#include <math.h>

#define BB 256
#define NN 100
#define DD 128
#define NP 112
#define HH 2
#define NITER 24

typedef __attribute__((ext_vector_type(2))) float f32x2;
typedef __attribute__((ext_vector_type(8))) float f32x8;

static __device__ __forceinline__ f32x8 wmma_f32(f32x2 a, f32x2 b, f32x8 c) {
  // D = A(16x4) x B(4x16) + C, f32 matrix pipe on gfx1250
  return __builtin_amdgcn_wmma_f32_16x16x4_f32(false, a, false, b, (short)0, c, false, false);
}

static __device__ __forceinline__ float wave_sum(float v) {
#pragma unroll
  for (int m = 16; m >= 1; m >>= 1) v += __shfl_xor(v, m, 32);
  return v;
}
static __device__ __forceinline__ float wave_max(float v) {
#pragma unroll
  for (int m = 16; m >= 1; m >>= 1) v = fmaxf(v, __shfl_xor(v, m, 32));
  return v;
}
static __device__ __forceinline__ float sigmoidf_(float x) {
  return 1.0f / (1.0f + __expf(-x));
}

// ---------------- prep kernels ----------------

__global__ __launch_bounds__(128) void k_prep1(
    const float* __restrict__ gnn, const unsigned char* __restrict__ m0,
    const unsigned char* __restrict__ m3, const float* __restrict__ fn_w,
    const float* __restrict__ fn_b, float* __restrict__ avg, float* __restrict__ dotA) {
  int b = blockIdx.x, d = threadIdx.x;
  int wave = d >> 5, lane = d & 31;
  float s = 0.f, cnt = 0.f;
  for (int n = 0; n < NN; ++n) {
    float mv = m0[b * NN + n] ? 1.f : 0.f;
    s += gnn[((size_t)b * NN + n) * DD + d] * mv;
    cnt += m3[b * NN + n] ? 1.f : 0.f;
  }
  float av = s / (cnt + 1e-7f);
  avg[b * DD + d] = av;
  __shared__ float red[4];
  float ws = wave_sum(av * fn_w[d]);
  if (lane == 0) red[wave] = ws;
  __syncthreads();
  if (d == 0) dotA[b] = red[0] + red[1] + red[2] + red[3] + fn_b[0];
}

__global__ __launch_bounds__(128) void k_prep2(
    const float* __restrict__ gnn, const float* __restrict__ drop, const float* __restrict__ proto,
    const unsigned char* __restrict__ m0, const unsigned char* __restrict__ m3,
    const float* __restrict__ fn_w, const float* __restrict__ dotA,
    const float* __restrict__ ssL_w, const float* __restrict__ ssL_b,
    const float* __restrict__ ssG_w, const float* __restrict__ ssG_b,
    float* __restrict__ gi, float* __restrict__ pi,
    float* __restrict__ aentL, float* __restrict__ aentG) {
  int bn = blockIdx.x, b = bn / NN, d = threadIdx.x;
  int wave = d >> 5, lane = d & 31;
  __shared__ float red[4];
  size_t base = (size_t)bn * DD;
  float m0v = m0[bn] ? 1.f : 0.f;
  float dmv = (!m0[bn] && m3[bn]) ? 1.f : 0.f;
  float di = drop[base + d] * dmv;
  float s = wave_sum(di * fn_w[DD + d]);
  if (lane == 0) red[wave] = s;
  __syncthreads();
  float alpha = dotA[b] + red[0] + red[1] + red[2] + red[3];
  float gv = gnn[base + d] * m0v + alpha * di;
  float pv = proto[base + d] * m0v;
  gi[base + d] = gv;
  pi[base + d] = pv;
  __syncthreads();
  float sL = wave_sum(gv * ssL_w[d]);
  if (lane == 0) red[wave] = sL;
  __syncthreads();
  if (d == 0) aentL[bn] = sigmoidf_(red[0] + red[1] + red[2] + red[3] + ssL_b[0]) + 1.0f;
  __syncthreads();
  float sG = wave_sum(pv * ssG_w[d]);
  if (lane == 0) red[wave] = sG;
  __syncthreads();
  if (d == 0) aentG[bn] = sigmoidf_(red[0] + red[1] + red[2] + red[3] + ssG_b[0]) + 1.0f;
}

// ---------------- attention (per batch x gnn) ----------------

__global__ __launch_bounds__(256) void k_attn(
    const float* __restrict__ gi, const float* __restrict__ pi,
    const float* __restrict__ aentL, const float* __restrict__ aentG,
    const float* __restrict__ laL_a, const float* __restrict__ laL_w, const float* __restrict__ laL_b,
    const float* __restrict__ laG_a, const float* __restrict__ laG_w, const float* __restrict__ laG_b,
    const int* __restrict__ adj, const unsigned char* __restrict__ m3,
    float* __restrict__ locb, float* __restrict__ glob,
    float* __restrict__ scS, float* __restrict__ outS) {
  const int blk = blockIdx.x;
  const int b = blk >> 1, g = blk & 1;
  const int tid = threadIdx.x, wave = tid >> 5, lane = tid & 31;
  const int lh = lane & 15;
  const int koff = (lane < 16) ? 0 : 2;

  const float* x = (g ? pi : gi) + (size_t)b * NN * DD;
  const float* aent = (g ? aentG : aentL) + (size_t)b * NN;
  const float* la_a = g ? laG_a : laL_a;
  const float* la_w = g ? laG_w : laL_w;
  const float* la_b = g ? laG_b : laL_b;
  float* outp = (g ? glob : locb) + (size_t)b * NN * DD;
  float* sc = scS + (size_t)blk * NP * NP;
  float* outs = outS + (size_t)blk * NP * DD;

  __shared__ float xs[NP * DD];   // 56KB: x tile, rows >= NN zero-padded
  __shared__ float avec[4 * DD];
  __shared__ float lbv[DD];

  for (int idx = tid; idx < NP * DD; idx += 256) {
    int r = idx >> 7;
    xs[idx] = (r < NN) ? x[(size_t)r * DD + (idx & 127)] : 0.f;
  }
  __syncthreads();

  for (int h = 0; h < HH; ++h) {
    for (int idx = tid; idx < 4 * DD; idx += 256) avec[idx] = la_a[(size_t)(h * 4) * DD + idx];
    for (int idx = tid; idx < DD; idx += 256) lbv[idx] = la_b[(size_t)h * DD + idx];
    __syncthreads();

    // phase 1: 4 edge-type logit GEMMs per tile, adj-select + leaky + col mask
    for (int t = wave; t < 49; t += 8) {
      int I = t / 7, J = t % 7;
      int m = I * 16 + lh, n = J * 16 + lh;
      f32x8 acc0 = {0,0,0,0,0,0,0,0}, acc1 = acc0, acc2 = acc0, acc3 = acc0;
      for (int kk = 0; kk < DD; kk += 4) {
        int k0 = kk + koff;
        float xa0 = xs[m * DD + k0], xa1 = xs[m * DD + k0 + 1];
        f32x2 bf; bf.x = xs[n * DD + k0]; bf.y = xs[n * DD + k0 + 1];
        f32x2 af;
        af.x = xa0 * avec[0 * DD + k0]; af.y = xa1 * avec[0 * DD + k0 + 1];
        acc0 = wmma_f32(af, bf, acc0);
        af.x = xa0 * avec[1 * DD + k0]; af.y = xa1 * avec[1 * DD + k0 + 1];
        acc1 = wmma_f32(af, bf, acc1);
        af.x = xa0 * avec[2 * DD + k0]; af.y = xa1 * avec[2 * DD + k0 + 1];
        acc2 = wmma_f32(af, bf, acc2);
        af.x = xa0 * avec[3 * DD + k0]; af.y = xa1 * avec[3 * DD + k0 + 1];
        acc3 = wmma_f32(af, bf, acc3);
      }
#pragma unroll
      for (int v = 0; v < 8; ++v) {
        int mm = I * 16 + v + ((lane < 16) ? 0 : 8);
        int nn = J * 16 + lh;
        int ad = (mm < NN && nn < NN) ? adj[((size_t)b * NN + mm) * NN + nn] : 0;
        float val = 0.f;
        if (ad == 1) val = acc0[v]; else if (ad == 2) val = acc1[v];
        else if (ad == 3) val = acc2[v]; else if (ad == 4) val = acc3[v];
        if (ad >= 1 && ad <= 4) val = (val >= 0.f) ? val : 0.2f * val;
        bool colok = (nn < NN) && (m3[(size_t)b * NN + nn] != 0);
        if (!colok) val = -INFINITY;
        sc[mm * NP + nn] = val;
      }
    }
    __threadfence();
    __syncthreads();

    // phase 2: entmax bisection, one wave per row, row in 4 regs/lane
    for (int i = wave; i < NN; i += 8) {
      float am1 = aent[i] - 1.0f;
      float inv = 1.0f / am1;
      float xr[4];
#pragma unroll
      for (int c = 0; c < 4; ++c) {
        int j = c * 32 + lane;
        xr[c] = (j < NP) ? sc[i * NP + j] * am1 : -INFINITY;
      }
      float mx = wave_max(fmaxf(fmaxf(xr[0], xr[1]), fmaxf(xr[2], xr[3])));
      float tau_lo = mx - 1.0f;
      float tau_hi = mx - __expf(am1 * __logf(1.0f / (float)NN));
      float s = 0.f;
#pragma unroll
      for (int c = 0; c < 4; ++c) {
        float z = xr[c] - tau_lo;
        if (z > 0.f) s += __expf(inv * __logf(z));
      }
      float f_lo = wave_sum(s) - 1.0f;
      float dm = tau_hi - tau_lo;
      float tau_m = tau_lo;
      for (int it = 0; it < NITER; ++it) {
        dm *= 0.5f;
        tau_m = tau_lo + dm;
        float sm = 0.f;
#pragma unroll
        for (int c = 0; c < 4; ++c) {
          float z = xr[c] - tau_m;
          if (z > 0.f) sm += __expf(inv * __logf(z));
        }
        float f_m = wave_sum(sm) - 1.0f;
        if (f_m * f_lo >= 0.f) tau_lo = tau_m;
      }
      float pm[4]; float ps = 0.f;
#pragma unroll
      for (int c = 0; c < 4; ++c) {
        float z = xr[c] - tau_m;
        pm[c] = (z > 0.f) ? __expf(inv * __logf(z)) : 0.f;
        ps += pm[c];
      }
      ps = wave_sum(ps);
      float rps = 1.0f / ps;
#pragma unroll
      for (int c = 0; c < 4; ++c) {
        int j = c * 32 + lane;
        if (j < NP) sc[i * NP + j] = pm[c] * rps;
      }
    }
    for (int i = NN + wave; i < NP; i += 8)
      for (int j = lane; j < NP; j += 32) sc[i * NP + j] = 0.f;
    __threadfence();
    __syncthreads();

    // phase 3: out = att @ x
    for (int t = wave; t < 56; t += 8) {
      int I = t / 8, J = t % 8;
      int m = I * 16 + lh, nd = J * 16 + lh;
      f32x8 acc = {0,0,0,0,0,0,0,0};
      for (int kk = 0; kk < NP; kk += 4) {
        int k0 = kk + koff;
        f32x2 af; af.x = sc[m * NP + k0]; af.y = sc[m * NP + k0 + 1];
        f32x2 bf; bf.x = xs[k0 * DD + nd]; bf.y = xs[(k0 + 1) * DD + nd];
        acc = wmma_f32(af, bf, acc);
      }
#pragma unroll
      for (int v = 0; v < 8; ++v) {
        int mm = I * 16 + v + ((lane < 16) ? 0 : 8);
        outs[mm * DD + nd] = acc[v];
      }
    }
    __threadfence();
    __syncthreads();

    // phase 4: gate = sigmoid([x|out] @ lw + lb); head out = g*out + (1-g)*x
    const float* lw = la_w + (size_t)h * 2 * DD * DD;
    for (int t = wave; t < 56; t += 8) {
      int I = t / 8, J = t % 8;
      int m = I * 16 + lh, nd = J * 16 + lh;
      f32x8 acc = {0,0,0,0,0,0,0,0};
      for (int kk = 0; kk < 2 * DD; kk += 4) {
        int k0 = kk + koff;
        f32x2 af;
        if (kk < DD) { af.x = xs[m * DD + k0];        af.y = xs[m * DD + k0 + 1]; }
        else         { af.x = outs[m * DD + k0 - DD]; af.y = outs[m * DD + k0 + 1 - DD]; }
        f32x2 bf; bf.x = lw[(size_t)k0 * DD + nd]; bf.y = lw[(size_t)(k0 + 1) * DD + nd];
        acc = wmma_f32(af, bf, acc);
      }
#pragma unroll
      for (int v = 0; v < 8; ++v) {
        int mm = I * 16 + v + ((lane < 16) ? 0 : 8);
        if (mm < NN) {
          float gt = sigmoidf_(acc[v] + lbv[nd]);
          float ov = outs[mm * DD + nd];
          float xv = xs[mm * DD + nd];
          float ho = gt * ov + (1.0f - gt) * xv;
          size_t o = (size_t)mm * DD + nd;
          if (h == 0) outp[o] = ho; else outp[o] += ho;
        }
      }
    }
    __threadfence();
    __syncthreads();
  }
}

// ---------------- (B*N x 128) @ (128 x 128) WMMA GEMM ----------------

__global__ __launch_bounds__(256) void k_gemm_nd(
    const float* __restrict__ in, const float* __restrict__ W, float* __restrict__ out) {
  int I = blockIdx.x;
  int tid = threadIdx.x, wave = tid >> 5, lane = tid & 31;
  int lh = lane & 15, koff = (lane < 16) ? 0 : 2;
  __shared__ float As[16 * DD];
  for (int idx = tid; idx < 16 * DD; idx += 256)
    As[idx] = in[(size_t)I * 16 * DD + idx];
  __syncthreads();
  int nd = wave * 16 + lh;
  f32x8 acc = {0,0,0,0,0,0,0,0};
  for (int kk = 0; kk < DD; kk += 4) {
    int k0 = kk + koff;
    f32x2 af; af.x = As[lh * DD + k0]; af.y = As[lh * DD + k0 + 1];
    f32x2 bf; bf.x = W[(size_t)k0 * DD + nd]; bf.y = W[(size_t)(k0 + 1) * DD + nd];
    acc = wmma_f32(af, bf, acc);
  }
#pragma unroll
  for (int v = 0; v < 8; ++v) {
    int mm = v + ((lane < 16) ? 0 : 8);
    out[((size_t)I * 16 + mm) * DD + nd] = acc[v];
  }
}

__global__ __launch_bounds__(128) void k_gem_combine(
    const float* __restrict__ U, const float* __restrict__ V,
    const float* __restrict__ loc, const float* __restrict__ glo,
    float* __restrict__ snew) {
  int r = blockIdx.x, d = threadIdx.x;
  int wave = d >> 5, lane = d & 31;
  __shared__ float red[4];
  size_t base = (size_t)r * DD;
  float u = U[base + d], v = V[base + d];
  float s = wave_sum(u * v);
  if (lane == 0) red[wave] = s;
  __syncthreads();
  float a = (red[0] + red[1] + red[2] + red[3]) * 0.08838834764831845f; // 1/sqrt(128)
  float l = loc[base + d], t = glo[base + d];
  snew[base + d] = l + a * (t - l);
}

// ---------------- final stage ----------------

__global__ __launch_bounds__(128) void k_gather_hs(
    const float* __restrict__ snew, const int* __restrict__ idxs,
    const unsigned char* __restrict__ m3, float* __restrict__ hid, float* __restrict__ hs) {
  int b = blockIdx.x, d = threadIdx.x;
  float acc = 0.f, cnt = 0.f;
  for (int i = 0; i < NN; ++i) {
    int j = idxs[b * NN + i];
    float mv = m3[b * NN + i] ? 1.f : 0.f;
    float v = snew[((size_t)b * NN + j) * DD + d] * mv;
    hid[((size_t)b * NN + i) * DD + d] = v;
    acc += v; cnt += mv;
  }
  hs[b * DD + d] = acc / (cnt + 1e-7f);
}

__global__ __launch_bounds__(128) void k_t2(
    const float* __restrict__ hs, const float* __restrict__ glu2_w, float* __restrict__ t2) {
  int b = blockIdx.x, e = threadIdx.x;
  float s = 0.f;
  for (int d = 0; d < DD; ++d) s += hs[b * DD + d] * glu2_w[d * DD + e];
  t2[b * DD + e] = s;
}

__global__ __launch_bounds__(256) void k_final(
    const float* __restrict__ hid, const float* __restrict__ pos,
    const float* __restrict__ w1, const float* __restrict__ glu1_w,
    const float* __restrict__ glu1_b, const float* __restrict__ t2,
    const float* __restrict__ w2, const unsigned char* __restrict__ m3,
    float* __restrict__ out) {
  int b = blockIdx.x;
  int tid = threadIdx.x, wave = tid >> 5, lane = tid & 31;
  int lh = lane & 15, koff = (lane < 16) ? 0 : 2;
  __shared__ float nh_s[16 * DD];
  __shared__ float beta_s[16];
  __shared__ float oacc[DD];
  if (tid < DD) oacc[tid] = 0.f;
  __syncthreads();
  for (int I = 0; I < 7; ++I) {
    {
      int m = I * 16 + lh;
      int nd = wave * 16 + lh;
      f32x8 acc = {0,0,0,0,0,0,0,0};
      for (int kk = 0; kk < 2 * DD; kk += 4) {
        int k0 = kk + koff;
        f32x2 af;
        if (m < NN) {
          if (kk < DD) { af.x = pos[(size_t)m * DD + k0]; af.y = pos[(size_t)m * DD + k0 + 1]; }
          else { af.x = hid[((size_t)b * NN + m) * DD + k0 - DD];
                 af.y = hid[((size_t)b * NN + m) * DD + k0 + 1 - DD]; }
        } else { af.x = 0.f; af.y = 0.f; }
        f32x2 bf; bf.x = w1[(size_t)k0 * DD + nd]; bf.y = w1[(size_t)(k0 + 1) * DD + nd];
        acc = wmma_f32(af, bf, acc);
      }
#pragma unroll
      for (int v = 0; v < 8; ++v) {
        int mm = v + ((lane < 16) ? 0 : 8);
        nh_s[mm * DD + nd] = tanhf(acc[v]);
      }
    }
    if (tid < 16) beta_s[tid] = 0.f;
    __syncthreads();
    {
      int nd = wave * 16 + lh;
      f32x8 acc = {0,0,0,0,0,0,0,0};
      for (int kk = 0; kk < DD; kk += 4) {
        int k0 = kk + koff;
        f32x2 af; af.x = nh_s[lh * DD + k0]; af.y = nh_s[lh * DD + k0 + 1];
        f32x2 bf; bf.x = glu1_w[(size_t)k0 * DD + nd]; bf.y = glu1_w[(size_t)(k0 + 1) * DD + nd];
        acc = wmma_f32(af, bf, acc);
      }
      float gb = glu1_b[nd] + t2[(size_t)b * DD + nd];
      float wv = w2[nd];
#pragma unroll
      for (int v = 0; v < 8; ++v) {
        int mm = v + ((lane < 16) ? 0 : 8);
        atomicAdd(&beta_s[mm], sigmoidf_(acc[v] + gb) * wv);
      }
    }
    __syncthreads();
    if (tid < DD) {
      for (int mm = 0; mm < 16; ++mm) {
        int i = I * 16 + mm;
        if (i < NN) {
          float bm = beta_s[mm] * (m3[b * NN + i] ? 1.f : 0.f);
          oacc[tid] += bm * hid[((size_t)b * NN + i) * DD + tid];
        }
      }
    }
    __syncthreads();
  }
  if (tid < DD) out[(size_t)b * DD + tid] = oacc[tid];
}

// ---------------- launcher ----------------

extern "C" void kernel_launch(void* const* d_in, const int* in_sizes, int n_in,
                              void* d_out, int out_size, void* d_ws, size_t ws_size,
                              hipStream_t stream) {
  const float* gnn    = (const float*)d_in[0];
  const float* drop   = (const float*)d_in[1];
  const float* proto  = (const float*)d_in[2];
  const float* pos    = (const float*)d_in[3];
  const float* laL_a  = (const float*)d_in[4];
  const float* laL_w  = (const float*)d_in[5];
  const float* laL_b  = (const float*)d_in[6];
  const float* ssL_w  = (const float*)d_in[7];
  const float* ssL_b  = (const float*)d_in[8];
  const float* laG_a  = (const float*)d_in[9];
  const float* laG_w  = (const float*)d_in[10];
  const float* laG_b  = (const float*)d_in[11];
  const float* ssG_w  = (const float*)d_in[12];
  const float* ssG_b  = (const float*)d_in[13];
  const float* gem_w  = (const float*)d_in[14];
  const float* fn_w   = (const float*)d_in[15];
  const float* fn_b   = (const float*)d_in[16];
  const float* w_1    = (const float*)d_in[17];
  const float* w_2    = (const float*)d_in[18];
  const float* glu1_w = (const float*)d_in[19];
  const float* glu1_b = (const float*)d_in[20];
  const float* glu2_w = (const float*)d_in[21];
  const int* adj          = (const int*)d_in[22];
  const int* input_index  = (const int*)d_in[23];
  const unsigned char* m0 = (const unsigned char*)d_in[24];
  const unsigned char* m3 = (const unsigned char*)d_in[25];
  float* out = (float*)d_out;

  float* ws = (float*)d_ws;
  const size_t BND = (size_t)BB * NN * DD;
  float* gi   = ws;               // later: U, then s_new
  float* pi   = ws + BND;         // later: V, then hid
  float* loc  = ws + 2 * BND;
  float* glo  = ws + 3 * BND;
  float* scS  = ws + 4 * BND;                         // 512 * 112*112
  float* outS = scS + (size_t)(2 * BB) * NP * NP;     // 512 * 112*128
  float* avg  = outS + (size_t)(2 * BB) * NP * DD;
  float* dotA = avg + (size_t)BB * DD;
  float* aentL = dotA + BB;
  float* aentG = aentL + (size_t)BB * NN;
  float* hs   = aentG + (size_t)BB * NN;
  float* t2   = hs + (size_t)BB * DD;

  k_prep1<<<BB, 128, 0, stream>>>(gnn, m0, m3, fn_w, fn_b, avg, dotA);
  k_prep2<<<BB * NN, 128, 0, stream>>>(gnn, drop, proto, m0, m3, fn_w, dotA,
                                       ssL_w, ssL_b, ssG_w, ssG_b, gi, pi, aentL, aentG);
  k_attn<<<BB * 2, 256, 0, stream>>>(gi, pi, aentL, aentG, laL_a, laL_w, laL_b,
                                     laG_a, laG_w, laG_b, adj, m3, loc, glo, scS, outS);
  k_gemm_nd<<<(BB * NN) / 16, 256, 0, stream>>>(loc, gem_w, gi);            // U = local @ w0
  k_gemm_nd<<<(BB * NN) / 16, 256, 0, stream>>>(glo, gem_w + DD * DD, pi);  // V = glob  @ w1
  k_gem_combine<<<BB * NN, 128, 0, stream>>>(gi, pi, loc, glo, gi);         // s_new -> gi
  k_gather_hs<<<BB, 128, 0, stream>>>(gi, input_index, m3, pi /*hid*/, hs);
  k_t2<<<BB, 128, 0, stream>>>(hs, glu2_w, t2);
  k_final<<<BB, 256, 0, stream>>>(pi /*hid*/, pos, w_1, glu1_w, glu1_b, t2, w_2, m3, out);
}